// LinkPredictionLoss_42863773614395
// MI455X (gfx1250) — compile-verified
//
#include <hip/hip_runtime.h>

#define N_PTS 4096
#define DIMS  1024
#define KNN   5
#define TILE  128
#define BK    16
#define LDS_STRIDE 20  // 80-byte rows: 16B-aligned b128 stores, conflict-free frag reads

typedef __attribute__((ext_vector_type(2))) float v2f;
typedef __attribute__((ext_vector_type(8))) float v8f;

#if __has_builtin(__builtin_amdgcn_global_load_async_to_lds_b128)
#define HAVE_ASYNC_LDS 1
typedef int v4i_vs __attribute__((vector_size(4 * sizeof(int))));
typedef __attribute__((address_space(1))) v4i_vs* as1_v4i_p;
typedef __attribute__((address_space(3))) v4i_vs* as3_v4i_p;
#else
#define HAVE_ASYNC_LDS 0
#endif

// 16-byte global -> LDS copy: async DMA path (ASYNCcnt-tracked) if available,
// otherwise synchronous b128 load + ds_store (compiler pipelines it).
__device__ __forceinline__ void copy16(const float* __restrict__ g, float* l) {
#if HAVE_ASYNC_LDS
    __builtin_amdgcn_global_load_async_to_lds_b128(
        (as1_v4i_p)(float*)g, (as3_v4i_p)l, 0, 0);
#else
    *(float4*)l = *(const float4*)g;
#endif
}

__device__ __forceinline__ void wait_async_copies() {
#if HAVE_ASYNC_LDS
#if __has_builtin(__builtin_amdgcn_s_wait_asynccnt)
    __builtin_amdgcn_s_wait_asynccnt(0);
#else
    asm volatile("s_wait_asynccnt 0" ::: "memory");
#endif
#endif
}

// ---------------------------------------------------------------- kernel 1
__global__ __launch_bounds__(256)
void sqnorm_kernel(const float* __restrict__ batch, float* __restrict__ sq) {
    const int row = blockIdx.x;
    const float* p = batch + (size_t)row * DIMS;
    float s = 0.0f;
    for (int c = threadIdx.x; c < DIMS; c += 256) {
        const float v = p[c];
        s += v * v;
    }
    __shared__ float red[256];
    red[threadIdx.x] = s;
    __syncthreads();
    for (int off = 128; off > 0; off >>= 1) {
        if (threadIdx.x < off) red[threadIdx.x] += red[threadIdx.x + off];
        __syncthreads();
    }
    if (threadIdx.x == 0) sq[row] = red[0];
}

// ---------------------------------------------------------------- kernel 2
// C = batch @ batch^T via V_WMMA_F32_16X16X4_F32, double-buffered LDS with
// async global->LDS staging, fused distance epilogue.
__global__ __launch_bounds__(256)
void dist_gemm_kernel(const float* __restrict__ batch,
                      const float* __restrict__ sq,
                      float* __restrict__ dmat) {
    __shared__ float As[2 * TILE * LDS_STRIDE];
    __shared__ float Bs[2 * TILE * LDS_STRIDE];

    const int tid   = threadIdx.x;
    const int lane  = tid & 31;
    const int wave  = tid >> 5;
    const int waveM = wave >> 2;        // 0..1 -> 64-row half
    const int waveN = wave & 3;         // 0..3 -> 32-col quarter
    const int lm    = lane & 15;        // A: M index / B: N index
    const int koff  = (lane >> 4) << 1; // K sub-offset: 0 or 2
    const int rowBase = blockIdx.y * TILE;
    const int colBase = blockIdx.x * TILE;

    v8f acc[4][2];
#pragma unroll
    for (int r = 0; r < 4; ++r)
#pragma unroll
        for (int c = 0; c < 2; ++c)
            acc[r][c] = (v8f){0.f, 0.f, 0.f, 0.f, 0.f, 0.f, 0.f, 0.f};

    // cooperative staging: 256 threads * 2 float4 per 128x16 slab (A and B)
    const int lr = tid >> 2;          // 0..63
    const int lc = (tid & 3) << 2;    // 0,4,8,12

    const float* gA0 = &batch[(size_t)(rowBase + lr) * DIMS + lc];
    const float* gA1 = &batch[(size_t)(rowBase + lr + 64) * DIMS + lc];
    const float* gB0 = &batch[(size_t)(colBase + lr) * DIMS + lc];
    const float* gB1 = &batch[(size_t)(colBase + lr + 64) * DIMS + lc];

#define STAGE(buf, kc)                                                        \
    do {                                                                      \
        const int o = (buf) * TILE * LDS_STRIDE;                              \
        copy16(gA0 + (kc), &As[o + lr * LDS_STRIDE + lc]);                    \
        copy16(gA1 + (kc), &As[o + (lr + 64) * LDS_STRIDE + lc]);             \
        copy16(gB0 + (kc), &Bs[o + lr * LDS_STRIDE + lc]);                    \
        copy16(gB1 + (kc), &Bs[o + (lr + 64) * LDS_STRIDE + lc]);             \
    } while (0)

    STAGE(0, 0);
    wait_async_copies();
    __syncthreads();

    int cur = 0;
    for (int kc = 0; kc < DIMS; kc += BK) {
        const int nxt = cur ^ 1;
        if (kc + BK < DIMS) STAGE(nxt, kc + BK);

        const float* a_base = &As[cur * TILE * LDS_STRIDE];
        const float* b_base = &Bs[cur * TILE * LDS_STRIDE];
#pragma unroll
        for (int kk = 0; kk < BK; kk += 4) {
            v2f a[4], b[2];
#pragma unroll
            for (int r = 0; r < 4; ++r) {
                const float* p =
                    &a_base[(waveM * 64 + r * 16 + lm) * LDS_STRIDE + kk + koff];
                a[r][0] = p[0];
                a[r][1] = p[1];
            }
#pragma unroll
            for (int c = 0; c < 2; ++c) {
                const float* p =
                    &b_base[(waveN * 32 + c * 16 + lm) * LDS_STRIDE + kk + koff];
                b[c][0] = p[0];
                b[c][1] = p[1];
            }
#pragma unroll
            for (int r = 0; r < 4; ++r)
#pragma unroll
                for (int c = 0; c < 2; ++c)
                    acc[r][c] = __builtin_amdgcn_wmma_f32_16x16x4_f32(
                        false, a[r], false, b[c], (short)0, acc[r][c],
                        false, false);
        }

        wait_async_copies();
        __syncthreads();
        cur = nxt;
    }
#undef STAGE

    // epilogue: d = max(sq[i] + sq[j] - 2*G, 0)
    const int mhalf = (lane >> 4) << 3; // 0 or 8
#pragma unroll
    for (int r = 0; r < 4; ++r) {
#pragma unroll
        for (int c = 0; c < 2; ++c) {
            const int gc = colBase + waveN * 32 + c * 16 + lm;
            const float sqc = sq[gc];
#pragma unroll
            for (int v = 0; v < 8; ++v) {
                const int gr = rowBase + waveM * 64 + r * 16 + v + mhalf;
                const float dist = sq[gr] + sqc - 2.0f * acc[r][c][v];
                dmat[(size_t)gr * N_PTS + gc] = fmaxf(dist, 0.0f);
            }
        }
    }
}

// ---------------------------------------------------------------- kernel 3
// per-row 5-NN selection (diagonal excluded) + log-softmax loss
__global__ __launch_bounds__(256)
void rowloss_kernel(const float* __restrict__ dmat,
                    const int* __restrict__ labels,
                    float* __restrict__ rowloss) {
    const int i   = blockIdx.x;
    const int tid = threadIdx.x;
    const float* row = dmat + (size_t)i * N_PTS;

    float vals[16];
    const int base = tid * 16;
#pragma unroll
    for (int t = 0; t < 16; ++t) vals[t] = row[base + t];

    __shared__ float sval[256];
    __shared__ int   sidx[256];
    __shared__ int   chosen[KNN];
    __shared__ float cdist[KNN];

    for (int k = 0; k < KNN; ++k) {
        float best = 3.0e38f;
        int   bidx = 0x7fffffff;
#pragma unroll
        for (int t = 0; t < 16; ++t) {
            const int j = base + t;
            bool skip = (j == i);
            for (int q = 0; q < k; ++q) skip = skip || (chosen[q] == j);
            const float v = vals[t];
            if (!skip && (v < best || (v == best && j < bidx))) {
                best = v;
                bidx = j;
            }
        }
        sval[tid] = best;
        sidx[tid] = bidx;
        __syncthreads();
        for (int off = 128; off > 0; off >>= 1) {
            if (tid < off) {
                const float ov = sval[tid + off];
                const int   oi = sidx[tid + off];
                if (ov < sval[tid] || (ov == sval[tid] && oi < sidx[tid])) {
                    sval[tid] = ov;
                    sidx[tid] = oi;
                }
            }
            __syncthreads();
        }
        if (tid == 0) {
            chosen[k] = sidx[0];
            cdist[k]  = sval[0];
        }
        __syncthreads();
    }

    if (tid == 0) {
        const int li = labels[i];
        float m = -cdist[0];
        for (int k = 1; k < KNN; ++k) m = fmaxf(m, -cdist[k]);
        float s = 0.0f;
        for (int k = 0; k < KNN; ++k) s += expf(-cdist[k] - m);
        const float lse = m + logf(s);
        float loss = 0.0f;
        for (int k = 0; k < KNN; ++k)
            if (labels[chosen[k]] == li) loss -= (-cdist[k] - lse);
        rowloss[i] = loss;
    }
}

// ---------------------------------------------------------------- kernel 4
__global__ __launch_bounds__(256)
void reduce_kernel(const float* __restrict__ rowloss, float* __restrict__ out) {
    __shared__ float red[256];
    float s = 0.0f;
    for (int idx = threadIdx.x; idx < N_PTS; idx += 256) s += rowloss[idx];
    red[threadIdx.x] = s;
    __syncthreads();
    for (int off = 128; off > 0; off >>= 1) {
        if (threadIdx.x < off) red[threadIdx.x] += red[threadIdx.x + off];
        __syncthreads();
    }
    if (threadIdx.x == 0) out[0] = red[0] * (1.0f / (float)N_PTS);
}

// ---------------------------------------------------------------- launcher
extern "C" void kernel_launch(void* const* d_in, const int* in_sizes, int n_in,
                              void* d_out, int out_size, void* d_ws,
                              size_t ws_size, hipStream_t stream) {
    const float* batch  = (const float*)d_in[0];
    const int*   labels = (const int*)d_in[1];
    float* out = (float*)d_out;

    float* sq      = (float*)d_ws;      // 4096 floats
    float* rowloss = sq + N_PTS;        // 4096 floats
    float* dmat    = rowloss + N_PTS;   // 4096*4096 floats (64 MB)

    sqnorm_kernel<<<N_PTS, 256, 0, stream>>>(batch, sq);

    dim3 grid(N_PTS / TILE, N_PTS / TILE);
    dist_gemm_kernel<<<grid, 256, 0, stream>>>(batch, sq, dmat);

    rowloss_kernel<<<N_PTS, 256, 0, stream>>>(dmat, labels, rowloss);
    reduce_kernel<<<1, 256, 0, stream>>>(rowloss, out);
}